// SuccessiveOverRelaxation_34892314312944
// MI455X (gfx1250) — compile-verified
//
#include <hip/hip_runtime.h>
#include <hip/hip_bf16.h>

// ---------------------------------------------------------------------------
// Red-black SOR Laplace solver (cortical-thickness style) for gfx1250.
// Padded volume: (bs*66, 66, 66) f32, strides: k=1, j=66, i=4356.
// Graph-friendly: fixed 501-iteration kernel chain with device-side
// convergence flag; TDM (tensor_load_to_lds) stages each block's contiguous
// row into LDS, fenced with s_wait_tensorcnt.
// ---------------------------------------------------------------------------

#define SOR_PI 3.14159265358979323846

struct Ctrl {
    unsigned converged;
    unsigned iters;
    float    dvLast;
};

typedef __attribute__((ext_vector_type(4))) unsigned int v4u;
typedef __attribute__((ext_vector_type(8))) int          v8i;
typedef __attribute__((ext_vector_type(4))) int          v4i;

// ---------------------------------------------------------------------------
// Init: build padded x0 (1 at sink label 3, else 0), color mask
// (0 fixed, 1 black = even parity GM, 2 red = odd parity GM), zero the
// guard regions, reset control block.
// ---------------------------------------------------------------------------
__global__ void sor_init(const float* __restrict__ img,
                         float* __restrict__ xg,          // guarded x base (xg[512] == cell 0)
                         unsigned char* __restrict__ mask,
                         Ctrl* __restrict__ ctrl,
                         int ntot) {
    long g = (long)blockIdx.x * blockDim.x + threadIdx.x;
    if (g == 0) {
        ctrl->converged = 0u;
        ctrl->iters = 0u;
        ctrl->dvLast = 0.0f;
    }
    if (g < 1024) {
        if (g < 512) xg[g] = 0.0f;                         // pre-guard
        else         xg[512 + ntot + (g - 512)] = 0.0f;    // post-guard
        return;
    }
    long idx = g - 1024;
    if (idx >= ntot) return;

    int k = (int)(idx % 66);
    int j = (int)((idx / 66) % 66);
    int I = (int)(idx / 4356);
    int b = I / 66;
    int i = I % 66;

    float lab = 0.0f;
    if (i >= 1 && i <= 64 && j >= 1 && j <= 64 && k >= 1 && k <= 64) {
        lab = img[(((long)b * 64 + (i - 1)) * 64 + (j - 1)) * 64 + (k - 1)];
    }
    float* x = xg + 512;
    x[idx] = (lab == 3.0f) ? 1.0f : 0.0f;

    unsigned char m = 0;
    if (lab == 1.0f) m = (unsigned char)(((I + j + k) & 1) ? 2 : 1); // parity0->black(1)
    mask[idx] = m;
}

// ---------------------------------------------------------------------------
// One colored half-sweep. Each block owns 256 contiguous cells; its
// [base-1, base+257) row is DMA'd into LDS by the Tensor Data Mover
// (wave 0 issues the descriptor; TENSORcnt fence; block barrier).
// Per-block |adj| sum goes to partial[blockIdx] (no atomics -> deterministic).
// ---------------------------------------------------------------------------
__global__ void __launch_bounds__(256)
sor_sweep(float* __restrict__ x,
          const unsigned char* __restrict__ mask,
          const Ctrl* __restrict__ ctrl,
          float* __restrict__ partial,
          int ntot, int colorSel, float wopt) {
    if (ctrl->converged) return;

    extern __shared__ float smem[];     // dynamic LDS at offset 0
    float* row    = smem;               // 258 floats: x[base-1 .. base+256]
    float* redbuf = smem + 272;         // 256 floats: reduction scratch

    const int t    = threadIdx.x;
    const int base = blockIdx.x * 256;

#if defined(__has_builtin) && __has_builtin(__builtin_amdgcn_tensor_load_to_lds)
    if (t < 32) {  // one wave issues the TDM descriptor (EXEC must be full for the wave)
        unsigned long long ga =
            (unsigned long long)(uintptr_t)(const void*)(x + base - 1);
        // D# group 0: count=1 | lds_addr=0 | global_addr | type=2 (bits 127:126)
        v4u g0 = { 1u,
                   0u,
                   (unsigned)ga,
                   (unsigned)((ga >> 32) & 0x01FFFFFFull) | 0x80000000u };
        // D# group 1: data_size=4B (enc 2); tensor_dim0=0x40000000 (no OOB clamp);
        // tensor_dim1=1; tile_dim0=258; tile_dim1=tile_dim2=0; stride0=0x40000000.
        v8i g1 = { 0x00020000, 0x00000000, 0x00014000, 0x01020000,
                   0x00000000, 0x40000000, 0x00000000, 0x00000000 };
        v4i gz4 = { 0, 0, 0, 0 };
        v8i gz8 = { 0, 0, 0, 0, 0, 0, 0, 0 };
        __builtin_amdgcn_tensor_load_to_lds(g0, g1, gz4, gz4, gz8, 0);
        __builtin_amdgcn_s_wait_tensorcnt(0);
    }
#else
    row[t] = x[base - 1 + t];
    if (t < 2) row[256 + t] = x[base + 255 + t];
#endif
    __syncthreads();

    const long idx = (long)base + t;
    float local = 0.0f;
    if (idx < ntot && mask[idx] == (unsigned char)colorSel) {
        float xc = row[t + 1];
        float s  = row[t] + row[t + 2]
                 + x[idx - 66]   + x[idx + 66]
                 + x[idx - 4356] + x[idx + 4356];
        float adj = wopt * (s - 6.0f * xc) * (1.0f / 6.0f);
        x[idx] = xc + adj;
        local = fabsf(adj);
    }

    redbuf[t] = local;
    __syncthreads();
    #pragma unroll
    for (int kk = 128; kk > 0; kk >>= 1) {
        if (t < kk) redbuf[t] += redbuf[t + kk];
        __syncthreads();
    }
    if (t == 0) partial[blockIdx.x] = redbuf[0];
}

// ---------------------------------------------------------------------------
// Convergence check: fixed-order reduction of black+red partials; records
// iters/dv of the just-finished body; flips the converged flag.
// ---------------------------------------------------------------------------
__global__ void sor_check(Ctrl* __restrict__ ctrl,
                          const float* __restrict__ pb,
                          const float* __restrict__ pr,
                          int nb, int it) {
    __shared__ float s[256];
    if (ctrl->converged) return;
    int t = threadIdx.x;
    float acc = 0.0f;
    for (int i = t; i < nb; i += 256) acc += pb[i] + pr[i];
    s[t] = acc;
    __syncthreads();
    for (int kk = 128; kk > 0; kk >>= 1) {
        if (t < kk) s[t] += s[t + kk];
        __syncthreads();
    }
    if (t == 0) {
        float dv = s[0];
        ctrl->iters  = (unsigned)it;
        ctrl->dvLast = dv;
        if (dv < 0.05f) ctrl->converged = 1u;
    }
}

// ---------------------------------------------------------------------------
// Crop the padded solution back to (bs,1,64,64,64); append iters and dv.
// ---------------------------------------------------------------------------
__global__ void sor_final(const float* __restrict__ x,
                          const Ctrl* __restrict__ ctrl,
                          float* __restrict__ out,
                          int outVol, int outSize) {
    long o = (long)blockIdx.x * blockDim.x + threadIdx.x;
    if (o < outVol) {
        int wk = (int)(o % 64);
        int hj = (int)((o / 64) % 64);
        int di = (int)((o / 4096) % 64);
        int b  = (int)(o / 262144);
        long xi = (((long)(b * 66 + di + 1)) * 66 + (hj + 1)) * 66 + (wk + 1);
        out[o] = x[xi];
    }
    if (o == 0) {
        if (outSize > outVol)     out[outVol]     = (float)ctrl->iters;
        if (outSize > outVol + 1) out[outVol + 1] = ctrl->dvLast;
    }
}

// ---------------------------------------------------------------------------
extern "C" void kernel_launch(void* const* d_in, const int* in_sizes, int n_in,
                              void* d_out, int out_size, void* d_ws, size_t ws_size,
                              hipStream_t stream) {
    const float* img = (const float*)d_in[0];
    float* out = (float*)d_out;
    char* ws = (char*)d_ws;

    const int bs     = in_sizes[0] / (64 * 64 * 64);   // (bs,1,64,64,64)
    const int ntot   = bs * 66 * 4356;                 // padded cells
    const int NB     = (ntot + 255) / 256;
    const int outVol = bs * 262144;

    // Workspace layout
    Ctrl* ctrl = (Ctrl*)ws;
    size_t off = 256;
    float* xg = (float*)(ws + off);              // 512 guard + ntot + 512 guard
    float* x  = xg + 512;
    off += (size_t)(ntot + 1024) * sizeof(float);
    unsigned char* mask = (unsigned char*)(ws + off);
    off += (size_t)ntot;
    off = (off + 255) & ~(size_t)255;
    float* pb = (float*)(ws + off);              // black partials
    off += (size_t)NB * sizeof(float);
    float* pr = (float*)(ws + off);              // red partials

    const float wopt = (float)(2.0 / (1.0 + SOR_PI / 66.0));
    const size_t shmem = (272 + 256) * sizeof(float);

    const int initThreads = ntot + 1024;
    sor_init<<<(initThreads + 255) / 256, 256, 0, stream>>>(img, xg, mask, ctrl, ntot);

    for (int it = 1; it <= 501; ++it) {
        sor_sweep<<<NB, 256, shmem, stream>>>(x, mask, ctrl, pb, ntot, 1, wopt); // black
        sor_sweep<<<NB, 256, shmem, stream>>>(x, mask, ctrl, pr, ntot, 2, wopt); // red
        sor_check<<<1, 256, 0, stream>>>(ctrl, pb, pr, NB, it);
    }

    sor_final<<<(outVol + 255) / 256, 256, 0, stream>>>(x, ctrl, out, outVol, out_size);
}